// SubgraphEmbeddingRegressorModel_75883482185879
// MI455X (gfx1250) — compile-verified
//
#include <hip/hip_runtime.h>

#define N_NODES   50000
#define N_EDGES   800000
#define N_GRAPHS  1024
#define N_PAIRS   16384
#define N_CL      16
#define IN_CH     64
#define EMB       128
#define HID       256

typedef float v2f __attribute__((ext_vector_type(2)));
typedef float v8f __attribute__((ext_vector_type(8)));

// ---------------- degree / normalization ----------------
__global__ __launch_bounds__(256) void k_deg_init(float* deg) {
  int i = blockIdx.x * 256 + threadIdx.x;
  if (i < N_NODES) deg[i] = 1.0f;           // self loop
}

__global__ __launch_bounds__(256) void k_deg_edges(const int* __restrict__ ei, float* deg) {
  int e = blockIdx.x * 256 + threadIdx.x;
  if (e < N_EDGES) atomicAdd(&deg[ei[N_EDGES + e]], 1.0f);  // target = edge_index[1]
}

__global__ __launch_bounds__(256) void k_dinv(const float* __restrict__ deg, float* dinv) {
  int i = blockIdx.x * 256 + threadIdx.x;
  if (i < N_NODES) dinv[i] = rsqrtf(deg[i]);
}

// ---------------- dense GEMM: C[M x N] = A[M x K] @ B[K x N] (fp32 WMMA) ----------
// grid = (M/16, N/64), block = 128 (4 waves, one 16x16 n-tile each).
// A tile staged in LDS (padded stride K+1 to kill bank conflicts).
__global__ __launch_bounds__(128) void k_gemm_wmma(const float* __restrict__ A,
                                                   const float* __restrict__ B,
                                                   float* __restrict__ C,
                                                   int K, int N) {
  __shared__ float As[16 * 129];            // K <= 128 here
  const int tid = threadIdx.x;
  const int m0  = blockIdx.x * 16;
  for (int idx = tid; idx < 16 * K; idx += 128) {
    int r = idx / K, c = idx - r * K;
    As[r * (K + 1) + c] = A[(size_t)(m0 + r) * K + c];
  }
  __syncthreads();

  const int wid  = tid >> 5, lane = tid & 31;
  const int half = lane >> 4, mrow = lane & 15;
  const int n0   = blockIdx.y * 64 + wid * 16;

  v8f acc = {};
  for (int k = 0; k < K; k += 4) {
    v2f a, b;
    // A 16x4 layout: lanes0-15 -> K={k,k+1}; lanes16-31 -> K={k+2,k+3}
    a.x = As[mrow * (K + 1) + k + 2 * half];
    a.y = As[mrow * (K + 1) + k + 1 + 2 * half];
    // B 4x16 layout: VGPR0 rows {k, k+2}, VGPR1 rows {k+1, k+3}, N striped on lanes
    b.x = B[(size_t)(k + 2 * half) * N + n0 + mrow];
    b.y = B[(size_t)(k + 1 + 2 * half) * N + n0 + mrow];
    acc = __builtin_amdgcn_wmma_f32_16x16x4_f32(false, a, false, b, (short)0, acc,
                                                false, false);
  }
  // C layout: VGPR i -> M = i + 8*half, N = mrow
#pragma unroll
  for (int i = 0; i < 8; ++i)
    C[(size_t)(m0 + i + 8 * half) * N + n0 + mrow] = acc[i];
}

// ---------------- GCN aggregation ----------------
__global__ __launch_bounds__(256) void k_agg_self(const float* __restrict__ dinv,
                                                  const float* __restrict__ hw,
                                                  float* __restrict__ agg) {
  int idx = blockIdx.x * 256 + threadIdx.x;
  if (idx < N_NODES * EMB) {
    float d = dinv[idx >> 7];
    agg[idx] = d * d * hw[idx];             // self-loop: norm = dinv[i]^2
  }
}

__global__ __launch_bounds__(256) void k_agg_edges(const int* __restrict__ ei,
                                                   const float* __restrict__ dinv,
                                                   const float* __restrict__ hw,
                                                   float* __restrict__ agg) {
  long long t = (long long)blockIdx.x * 256 + threadIdx.x;
  if (t >= (long long)N_EDGES * 32) return; // 32 threads/edge, 4 floats each
  int e = (int)(t >> 5);
  int f = (int)(t & 31) * 4;
  int row = ei[e], col = ei[N_EDGES + e];
  float w = dinv[row] * dinv[col];
  const float4 v = *(const float4*)(hw + (size_t)row * EMB + f);
  float* dst = agg + (size_t)col * EMB + f;
  atomicAdd(dst + 0, w * v.x);
  atomicAdd(dst + 1, w * v.y);
  atomicAdd(dst + 2, w * v.z);
  atomicAdd(dst + 3, w * v.w);
}

__global__ __launch_bounds__(256) void k_bias_relu(float* __restrict__ h,
                                                   const float* __restrict__ b) {
  int idx = blockIdx.x * 256 + threadIdx.x;
  if (idx < N_NODES * EMB) {
    float v = h[idx] + b[idx & (EMB - 1)];
    h[idx] = v > 0.f ? v : 0.f;
  }
}

// ---------------- scatter-mean pooling ----------------
__global__ __launch_bounds__(256) void k_pool_zero(float* gsum, float* cnt) {
  int idx = blockIdx.x * 256 + threadIdx.x;
  if (idx < N_GRAPHS * EMB) gsum[idx] = 0.f;
  if (idx < N_GRAPHS) cnt[idx] = 0.f;
}

__global__ __launch_bounds__(256) void k_pool_accum(const int* __restrict__ batch,
                                                    const float* __restrict__ h,
                                                    float* gsum, float* cnt) {
  int idx = blockIdx.x * 256 + threadIdx.x;
  if (idx >= N_NODES * EMB) return;
  int i = idx >> 7, f = idx & (EMB - 1);
  int g = batch[i];
  atomicAdd(&gsum[(size_t)g * EMB + f], h[idx]);
  if (f == 0) atomicAdd(&cnt[g], 1.0f);
}

__global__ __launch_bounds__(256) void k_pool_div(const float* __restrict__ gsum,
                                                  const float* __restrict__ cnt,
                                                  float* __restrict__ gemb) {
  int idx = blockIdx.x * 256 + threadIdx.x;
  if (idx < N_GRAPHS * EMB)
    gemb[idx] = gsum[idx] / fmaxf(cnt[idx >> 7], 1.0f);
}

// ---------------- bucket pairs by cell line ----------------
__global__ __launch_bounds__(64) void k_bucket_zero(int* cnt_c, int* cursor) {
  int t = threadIdx.x;
  if (t < N_CL) { cnt_c[t] = 0; cursor[t] = 0; }
}

__global__ __launch_bounds__(256) void k_bucket_count(const int* __restrict__ cell,
                                                      int* cnt_c) {
  int p = blockIdx.x * 256 + threadIdx.x;
  if (p < N_PAIRS) atomicAdd(&cnt_c[cell[p]], 1);
}

__global__ void k_prefix(const int* __restrict__ cnt_c, int* off_c, int* tileoff) {
  off_c[0] = 0; tileoff[0] = 0;
  for (int c = 0; c < N_CL; ++c) {
    off_c[c + 1]   = off_c[c] + cnt_c[c];
    tileoff[c + 1] = tileoff[c] + (cnt_c[c] + 15) / 16;
  }
}

__global__ __launch_bounds__(256) void k_bucket_scatter(const int* __restrict__ cell,
                                                        const int* __restrict__ off_c,
                                                        int* cursor, int* perm) {
  int p = blockIdx.x * 256 + threadIdx.x;
  if (p >= N_PAIRS) return;
  int c = cell[p];
  int pos = atomicAdd(&cursor[c], 1);
  perm[off_c[c] + pos] = p;
}

// ---------------- regressor layer 1: gathered WMMA GEMM, fused bias+ReLU ----------
// grid = (maxTiles, HID/64), block = 128.  A rows gathered from graph embeds
// via pair buckets so every tile shares one W1[c].
__global__ __launch_bounds__(128) void k_reg_gemm1(const float* __restrict__ G,
                                                   const int* __restrict__ ddb,
                                                   const float* __restrict__ W1,
                                                   const float* __restrict__ B1,
                                                   const int* __restrict__ perm,
                                                   const int* __restrict__ off_c,
                                                   const int* __restrict__ cnt_c,
                                                   const int* __restrict__ tileoff,
                                                   float* __restrict__ hidden) {
  __shared__ float As[16 * 257];
  __shared__ int   sp[16];
  const int t = blockIdx.x;
  if (t >= tileoff[N_CL]) return;           // block-uniform
  int c = 0;
#pragma unroll
  for (int cc = 0; cc < N_CL; ++cc)
    if (t >= tileoff[cc]) c = cc;
  const int lt  = t - tileoff[c];
  const int tid = threadIdx.x;
  if (tid < 16) {
    int r = lt * 16 + tid;
    sp[tid] = (r < cnt_c[c]) ? perm[off_c[c] + r] : -1;
  }
  __syncthreads();
  // stage gathered pair tile [16 x 256] : concat(G[q0], G[q1])
  for (int idx = tid; idx < 16 * 256; idx += 128) {
    int r = idx >> 8, e = idx & 255;
    int p = sp[r];
    float v = 0.f;
    if (p >= 0) {
      int g = (e < EMB) ? ddb[p] : ddb[N_PAIRS + p];
      v = G[(size_t)g * EMB + (e & (EMB - 1))];
    }
    As[r * 257 + e] = v;
  }
  __syncthreads();

  const int wid  = tid >> 5, lane = tid & 31;
  const int half = lane >> 4, mrow = lane & 15;
  const int n0   = blockIdx.y * 64 + wid * 16;
  const float* Bp = W1 + (size_t)c * (2 * EMB) * HID;

  v8f acc = {};
  for (int k = 0; k < 2 * EMB; k += 4) {
    v2f a, b;
    a.x = As[mrow * 257 + k + 2 * half];
    a.y = As[mrow * 257 + k + 1 + 2 * half];
    b.x = Bp[(size_t)(k + 2 * half) * HID + n0 + mrow];
    b.y = Bp[(size_t)(k + 1 + 2 * half) * HID + n0 + mrow];
    acc = __builtin_amdgcn_wmma_f32_16x16x4_f32(false, a, false, b, (short)0, acc,
                                                false, false);
  }
  const float bias = B1[c * HID + n0 + mrow];
#pragma unroll
  for (int i = 0; i < 8; ++i) {
    int p = sp[i + 8 * half];
    if (p >= 0) {
      float v = acc[i] + bias;
      hidden[(size_t)p * HID + n0 + mrow] = v > 0.f ? v : 0.f;
    }
  }
}

// ---------------- regressor layer 2: 256 -> 1 dot, one wave per pair ------------
__global__ __launch_bounds__(256) void k_reg_dot(const float* __restrict__ hidden,
                                                 const float* __restrict__ W2,
                                                 const float* __restrict__ B2,
                                                 const int* __restrict__ cell,
                                                 float* __restrict__ out) {
  int p = blockIdx.x * 8 + (threadIdx.x >> 5);
  int lane = threadIdx.x & 31;
  if (p >= N_PAIRS) return;
  int c = cell[p];
  float s = 0.f;
#pragma unroll
  for (int j = 0; j < 8; ++j) {
    int e = lane + j * 32;
    s += hidden[(size_t)p * HID + e] * W2[c * HID + e];
  }
  for (int o = 16; o > 0; o >>= 1) s += __shfl_xor(s, o, 32);
  if (lane == 0) out[p] = s + B2[c];
}

// =======================================================================
extern "C" void kernel_launch(void* const* d_in, const int* in_sizes, int n_in,
                              void* d_out, int out_size, void* d_ws, size_t ws_size,
                              hipStream_t stream) {
  const float* x    = (const float*)d_in[0];
  const float* w1   = (const float*)d_in[1];
  const float* b1   = (const float*)d_in[2];
  const float* w2   = (const float*)d_in[3];
  const float* b2   = (const float*)d_in[4];
  const float* rw1  = (const float*)d_in[5];
  const float* rb1  = (const float*)d_in[6];
  const float* rw2  = (const float*)d_in[7];
  const float* rb2  = (const float*)d_in[8];
  const int*   ei   = (const int*)d_in[9];
  const int*   batch= (const int*)d_in[10];
  const int*   ddb  = (const int*)d_in[11];   // q == ddb (searchsorted identity)
  const int*   cell = (const int*)d_in[12];
  float*       out  = (float*)d_out;

  char* ws = (char*)d_ws;
  size_t o = 0;
  auto alloc = [&](size_t bytes) -> void* {
    void* p = ws + o;
    o = (o + bytes + 255) & ~(size_t)255;
    return p;
  };
  float* buf0  = (float*)alloc((size_t)N_NODES * EMB * 4);   // hw
  float* buf1  = (float*)alloc((size_t)N_NODES * EMB * 4);   // agg / h
  float* deg   = (float*)alloc((size_t)N_NODES * 4);
  float* dinv  = (float*)alloc((size_t)N_NODES * 4);
  float* gsum  = (float*)alloc((size_t)N_GRAPHS * EMB * 4);
  float* cntg  = (float*)alloc((size_t)N_GRAPHS * 4);
  float* gemb  = (float*)alloc((size_t)N_GRAPHS * EMB * 4);
  float* hidden= (float*)alloc((size_t)N_PAIRS * HID * 4);
  int*   cnt_c = (int*)alloc(16 * 4);
  int*   off_c = (int*)alloc(17 * 4);
  int*   tilef = (int*)alloc(17 * 4);
  int*   cursor= (int*)alloc(16 * 4);
  int*   perm  = (int*)alloc((size_t)N_PAIRS * 4);

  // --- normalization ---
  k_deg_init <<<(N_NODES + 255) / 256, 256, 0, stream>>>(deg);
  k_deg_edges<<<(N_EDGES + 255) / 256, 256, 0, stream>>>(ei, deg);
  k_dinv     <<<(N_NODES + 255) / 256, 256, 0, stream>>>(deg, dinv);

  // --- GCN layer 1 ---
  k_gemm_wmma<<<dim3(N_NODES / 16, EMB / 64), 128, 0, stream>>>(x, w1, buf0, IN_CH, EMB);
  k_agg_self <<<(N_NODES * EMB + 255) / 256, 256, 0, stream>>>(dinv, buf0, buf1);
  k_agg_edges<<<(int)(((long long)N_EDGES * 32 + 255) / 256), 256, 0, stream>>>(ei, dinv, buf0, buf1);
  k_bias_relu<<<(N_NODES * EMB + 255) / 256, 256, 0, stream>>>(buf1, b1);

  // --- GCN layer 2 ---
  k_gemm_wmma<<<dim3(N_NODES / 16, EMB / 64), 128, 0, stream>>>(buf1, w2, buf0, EMB, EMB);
  k_agg_self <<<(N_NODES * EMB + 255) / 256, 256, 0, stream>>>(dinv, buf0, buf1);
  k_agg_edges<<<(int)(((long long)N_EDGES * 32 + 255) / 256), 256, 0, stream>>>(ei, dinv, buf0, buf1);
  k_bias_relu<<<(N_NODES * EMB + 255) / 256, 256, 0, stream>>>(buf1, b2);

  // --- pooling ---
  k_pool_zero <<<(N_GRAPHS * EMB + 255) / 256, 256, 0, stream>>>(gsum, cntg);
  k_pool_accum<<<(N_NODES * EMB + 255) / 256, 256, 0, stream>>>(batch, buf1, gsum, cntg);
  k_pool_div  <<<(N_GRAPHS * EMB + 255) / 256, 256, 0, stream>>>(gsum, cntg, gemb);

  // --- bucket pairs by cell line ---
  k_bucket_zero   <<<1, 64, 0, stream>>>(cnt_c, cursor);
  k_bucket_count  <<<(N_PAIRS + 255) / 256, 256, 0, stream>>>(cell, cnt_c);
  k_prefix        <<<1, 1, 0, stream>>>(cnt_c, off_c, tilef);
  k_bucket_scatter<<<(N_PAIRS + 255) / 256, 256, 0, stream>>>(cell, off_c, cursor, perm);

  // --- regressor (only selected cell line per pair) ---
  const int maxTiles = N_PAIRS / 16 + N_CL;  // 1040 upper bound
  k_reg_gemm1<<<dim3(maxTiles, HID / 64), 128, 0, stream>>>(gemb, ddb, rw1, rb1,
                                                            perm, off_c, cnt_c, tilef, hidden);
  k_reg_dot  <<<N_PAIRS / 8, 256, 0, stream>>>(hidden, rw2, rb2, cell, out);

  (void)in_sizes; (void)n_in; (void)out_size; (void)ws_size;
}